// STC_layer_22428319220734
// MI455X (gfx1250) — compile-verified
//
#include <hip/hip_runtime.h>
#include <math.h>

typedef __attribute__((ext_vector_type(2))) float v2f;
typedef __attribute__((ext_vector_type(4))) float v4f;
typedef __attribute__((ext_vector_type(8))) float v8f;

#define FDIM 128
#define FS   12
#define NS   10   // num_sample = F-2

// ---------------------------------------------------------------------------
// Kernel 1 (single wave): eigh of the star-Laplacian filter (double Jacobi),
// then v = U * diag(w) * U^T * avgweight via V_WMMA_F32_16X16X4_F32 (F padded
// to 16).  Writes coeff[k] = v[k+1], k=0..9, into d_ws.
// ---------------------------------------------------------------------------
__global__ __launch_bounds__(32) void spectral_coeff_kernel(
    const float* __restrict__ weight,
    const float* __restrict__ avgweight,
    float* __restrict__ coeff)
{
    __shared__ double sA[FS][FS];
    __shared__ double sV[FS][FS];
    __shared__ float  sU[16][16];   // eigenbasis, zero padded
    __shared__ float  sT[16][16];   // T = U * diag(w)
    __shared__ float  sM[16][16];   // M = T * U^T
    __shared__ float  sW[16];       // weight, zero padded

    const int lane = threadIdx.x;

    // stage weight into padded LDS (removes divergent global loads around WMMA)
    if (lane < 16) sW[lane] = (lane < FS) ? weight[lane] : 0.0f;

    if (lane == 0) {
        // filt = I - D A D for the star graph (F=12)
        const double invs = 1.0 / sqrt((double)(FS - 1));
        for (int i = 0; i < FS; ++i)
            for (int j = 0; j < FS; ++j) {
                double a = (i == j) ? 1.0 : 0.0;
                if (i == 0 && j > 0) a = -invs;
                if (j == 0 && i > 0) a = -invs;
                sA[i][j] = a;
                sV[i][j] = (i == j) ? 1.0 : 0.0;
            }
        // cyclic Jacobi eigensolver (double precision, deterministic)
        for (int sweep = 0; sweep < 64; ++sweep) {
            double off = 0.0;
            for (int p = 0; p < FS; ++p)
                for (int q = p + 1; q < FS; ++q) off += sA[p][q] * sA[p][q];
            if (off < 1e-28) break;
            for (int p = 0; p < FS - 1; ++p)
                for (int q = p + 1; q < FS; ++q) {
                    double apq = sA[p][q];
                    if (fabs(apq) < 1e-300) continue;
                    double theta = 0.5 * (sA[q][q] - sA[p][p]) / apq;
                    double t = ((theta >= 0.0) ? 1.0 : -1.0) /
                               (fabs(theta) + sqrt(theta * theta + 1.0));
                    double c = 1.0 / sqrt(t * t + 1.0);
                    double s = t * c;
                    for (int k = 0; k < FS; ++k) {       // column rotation
                        double akp = sA[k][p], akq = sA[k][q];
                        sA[k][p] = c * akp - s * akq;
                        sA[k][q] = s * akp + c * akq;
                    }
                    for (int k = 0; k < FS; ++k) {       // row rotation
                        double apk = sA[p][k], aqk = sA[q][k];
                        sA[p][k] = c * apk - s * aqk;
                        sA[q][k] = s * apk + c * aqk;
                    }
                    for (int k = 0; k < FS; ++k) {       // accumulate vectors
                        double vkp = sV[k][p], vkq = sV[k][q];
                        sV[k][p] = c * vkp - s * vkq;
                        sV[k][q] = s * vkp + c * vkq;
                    }
                }
        }
        // sort eigenvalues ascending (eigh convention), permute columns
        int perm[FS];
        double ev[FS];
        for (int i = 0; i < FS; ++i) { perm[i] = i; ev[i] = sA[i][i]; }
        for (int i = 0; i < FS - 1; ++i) {
            int mi = i;
            for (int j = i + 1; j < FS; ++j)
                if (ev[perm[j]] < ev[perm[mi]]) mi = j;
            int tp = perm[i]; perm[i] = perm[mi]; perm[mi] = tp;
        }
        for (int r = 0; r < 16; ++r)
            for (int cc = 0; cc < 16; ++cc)
                sU[r][cc] = (r < FS && cc < FS) ? (float)sV[r][perm[cc]] : 0.0f;
    }
    __syncthreads();

    // ---- WMMA stage (whole wave, EXEC all ones) ----
    // A 16x4 f32 layout: lanes 0-15 row M=lane&15, K={0,1}; lanes 16-31 K={2,3}.
    // C/D 16x16 f32 layout: VGPR j -> M=j (lanes 0-15) / M=j+8 (lanes 16-31).
    const int n  = lane & 15;          // A row / B,C column
    const int hi = (lane >> 4) & 1;    // K-half selector

    // Stage 1: T = U * diag(w)   (K=16 via 4 chained 16x16x4 WMMAs)
    v8f acc = {0.f, 0.f, 0.f, 0.f, 0.f, 0.f, 0.f, 0.f};
#pragma unroll
    for (int kk = 0; kk < 4; ++kk) {
        const int k0 = kk * 4 + hi * 2;
        v2f a, b;
        a.x = sU[n][k0 + 0];
        a.y = sU[n][k0 + 1];
        b.x = ((k0 + 0) == n) ? sW[n] : 0.0f;   // diag(w), padded
        b.y = ((k0 + 1) == n) ? sW[n] : 0.0f;
        acc = __builtin_amdgcn_wmma_f32_16x16x4_f32(
            false, a, false, b, (short)0, acc, false, false);
    }
#pragma unroll
    for (int j = 0; j < 8; ++j) sT[j + hi * 8][n] = acc[j];
    __syncthreads();

    // Stage 2: M = T * U^T   (B[k][n] = U[n][k])
    v8f accM = {0.f, 0.f, 0.f, 0.f, 0.f, 0.f, 0.f, 0.f};
#pragma unroll
    for (int kk = 0; kk < 4; ++kk) {
        const int k0 = kk * 4 + hi * 2;
        v2f a, b;
        a.x = sT[n][k0 + 0];
        a.y = sT[n][k0 + 1];
        b.x = sU[n][k0 + 0];
        b.y = sU[n][k0 + 1];
        accM = __builtin_amdgcn_wmma_f32_16x16x4_f32(
            false, a, false, b, (short)0, accM, false, false);
    }
#pragma unroll
    for (int j = 0; j < 8; ++j) sM[j + hi * 8][n] = accM[j];
    __syncthreads();

    // coeff[k] = (M * avgweight)[k+1]  (rows 0 and F-1 of mask2 are zero)
    if (lane == 0) {
        for (int k = 0; k < NS; ++k) {
            float s = 0.f;
            for (int j = 0; j < FS; ++j) s += sM[k + 1][j] * avgweight[j];
            coeff[k] = s;
        }
    }
}

// ---------------------------------------------------------------------------
// Kernel 2: bandwidth-bound gather.  One wave per output row b:
// out[b,:] = sum_k coeff[k] * E[idx[b,k], :].  Each lane owns 16B (b128),
// so each of the 10 gathers is one coalesced 512B row read.  Embedding reads
// stay temporal (table ~fits 192MB L2, ~1.25x reuse); index reads and the
// output store are non-temporal so the touch-once streams don't evict it.
// ---------------------------------------------------------------------------
__global__ __launch_bounds__(256) void gather_wsum_kernel(
    const float* __restrict__ E,
    const int*   __restrict__ idx,
    const float* __restrict__ coeff,
    float*       __restrict__ out,
    int B)
{
    const int wave = threadIdx.x >> 5;
    const int lane = threadIdx.x & 31;
    const int b = blockIdx.x * 8 + wave;
    if (b >= B) return;

    float c[NS];
#pragma unroll
    for (int k = 0; k < NS; ++k) c[k] = coeff[k];   // wave-uniform -> scalar

    const int base = b * NS;
    v4f acc = {0.f, 0.f, 0.f, 0.f};
#pragma unroll
    for (int k = 0; k < NS; ++k) {
        const int row = __builtin_nontemporal_load(idx + base + k);
        const v4f e = *((const v4f*)(E + (size_t)row * FDIM) + lane);
        acc += c[k] * e;
    }
    __builtin_nontemporal_store(acc, (v4f*)(out + (size_t)b * FDIM) + lane);
}

// ---------------------------------------------------------------------------
extern "C" void kernel_launch(void* const* d_in, const int* in_sizes, int n_in,
                              void* d_out, int out_size, void* d_ws, size_t ws_size,
                              hipStream_t stream) {
    const float* E         = (const float*)d_in[0];  // [N,128]
    const float* weight    = (const float*)d_in[1];  // [12,1]
    const float* avgweight = (const float*)d_in[2];  // [12,1]
    const int*   idx       = (const int*)d_in[3];    // [B,10]
    float* coeff = (float*)d_ws;                     // 10 floats scratch
    const int B = in_sizes[3] / NS;

    spectral_coeff_kernel<<<1, 32, 0, stream>>>(weight, avgweight, coeff);
    gather_wsum_kernel<<<(B + 7) / 8, 256, 0, stream>>>(
        E, idx, coeff, (float*)d_out, B);
}